// Adaptive_dilatedConv_86827058856254
// MI455X (gfx1250) — compile-verified
//
#include <hip/hip_runtime.h>
#include <math.h>

// ---------------------------------------------------------------------------
// Adaptive dilated conv block for MI455X (gfx1250, wave32).
//   1) attn_kernel : per-pixel softmax over 3 branch logits -> scale planes
//   2) fused_kernel: 3 dilated depthwise 3x3 convs combined with scales
//   3) gemm_kernel : out 1x1 conv as fp32 WMMA GEMM (v_wmma_f32_16x16x4_f32)
//                    with double-buffered GLOBAL_LOAD_ASYNC_TO_LDS_B128 staging
// Workspace layout: [scale: 3*65536 f32][fused: 16*256*4096 f32]
// ---------------------------------------------------------------------------

#define CCH    256
#define HW     4096
#define NPIX   65536      // B * HW
#define KSTAGE 64
#define MBLK   128
#define NBLK   64
#define KSTR   68         // 64 + 4 pad: rows stay 16B aligned for async B128

typedef __attribute__((ext_vector_type(2))) float v2f;
typedef __attribute__((ext_vector_type(8))) float v8f;

// Per-lane async DMA: copy 16B global -> 16B LDS, tracked by ASYNCcnt.
__device__ __forceinline__ void async_ld_b128(unsigned lds_off, unsigned gvoff,
                                              const float* sbase) {
    asm volatile("global_load_async_to_lds_b128 %0, %1, %2"
                 :: "v"(lds_off), "v"(gvoff), "s"(sbase)
                 : "memory");
}
__device__ __forceinline__ void wait_async0() {
    asm volatile("s_wait_asynccnt 0x0" ::: "memory");
}
__device__ __forceinline__ unsigned lds_off32(const void* p) {
    // LDS aperture: generic addr low 32 bits == LDS byte offset (ISA 10.2)
    return (unsigned)(size_t)p;
}

// ---------------------------------------------------------------- kernel 1 --
__global__ __launch_bounds__(256)
void attn_kernel(const float* __restrict__ x,
                 const float* __restrict__ attn_w,   // [3, C]
                 const float* __restrict__ attn_b,   // [3]
                 float* __restrict__ scale) {        // [3, NPIX]
    int g = blockIdx.x * 256 + threadIdx.x;          // global pixel 0..65535
    int b = g >> 12;
    int p = g & (HW - 1);
    const float* xp = x + (size_t)b * CCH * HW + p;

    float l0 = 0.f, l1 = 0.f, l2 = 0.f;
#pragma unroll 4
    for (int c = 0; c < CCH; ++c) {
        float xv = xp[c * HW];                       // coalesced along w
        l0 = fmaf(xv, attn_w[c], l0);                // uniform -> s_load
        l1 = fmaf(xv, attn_w[CCH + c], l1);
        l2 = fmaf(xv, attn_w[2 * CCH + c], l2);
    }
    l0 += attn_b[0]; l1 += attn_b[1]; l2 += attn_b[2];
    float m  = fmaxf(l0, fmaxf(l1, l2));
    float e0 = expf(l0 - m), e1 = expf(l1 - m), e2 = expf(l2 - m);
    float inv = 1.0f / (e0 + e1 + e2);
    scale[g]             = fmaf(e0, inv, 1.0f);      // attn + 1
    scale[NPIX + g]      = fmaf(e1, inv, 1.0f);
    scale[2 * NPIX + g]  = fmaf(e2, inv, 1.0f);
}

// ---------------------------------------------------------------- kernel 2 --
__global__ __launch_bounds__(256)
void fused_kernel(const float* __restrict__ x,
                  const float* __restrict__ dw_w,    // [3, C, 1, 3, 3]
                  const float* __restrict__ dw_b,    // [3, C]
                  const float* __restrict__ scale,   // [3, NPIX]
                  float* __restrict__ fused) {       // [B, C, HW]
    const int DIL[3] = {1, 2, 5};
    int bc = blockIdx.x;                             // 0 .. B*C-1
    int b  = bc >> 8;
    int c  = bc & (CCH - 1);
    const float* xp = x + (size_t)bc * HW;

    float wk[3][9], bk[3];
#pragma unroll
    for (int k = 0; k < 3; ++k) {
        bk[k] = dw_b[k * CCH + c];                   // uniform -> SGPR
#pragma unroll
        for (int t = 0; t < 9; ++t)
            wk[k][t] = dw_w[(size_t)(k * CCH + c) * 9 + t];
    }

    for (int p = threadIdx.x; p < HW; p += 256) {
        int h = p >> 6, w = p & 63;
        float conv[3];
#pragma unroll
        for (int k = 0; k < 3; ++k) {
            int d = DIL[k];
            float s = 0.f;
#pragma unroll
            for (int i = 0; i < 3; ++i) {
#pragma unroll
                for (int j = 0; j < 3; ++j) {
                    int hh = h + (i - 1) * d;
                    int ww = w + (j - 1) * d;
                    float xv = ((unsigned)hh < 64u && (unsigned)ww < 64u)
                                   ? xp[hh * 64 + ww] : 0.f;
                    s = fmaf(wk[k][i * 3 + j], xv, s);
                }
            }
            conv[k] = s + bk[k];
        }
        int g = b * HW + p;
        float r = scale[g] * conv[0]
                + scale[NPIX + g] * conv[1]
                + scale[2 * NPIX + g] * conv[2];
        fused[(size_t)bc * HW + p] = r;
    }
}

// ---------------------------------------------------------------- kernel 3 --
// out[b, o, p] = sum_c out_w[o,c] * fused[b,c,p] + out_b[o]
// Block tile: M=128 x N=64; K staged by 64 through double-buffered LDS filled
// by per-lane async DMA (global_load_async_to_lds_b128, ASYNCcnt-tracked).
// 8 waves; wave w: N-tile = w%4, M-group = w/4 (4 M-tiles of 16 each).
__global__ __launch_bounds__(256)
void gemm_kernel(const float* __restrict__ fused,   // [B, C, HW]
                 const float* __restrict__ out_w,   // [256, 256] row-major
                 const float* __restrict__ out_b,   // [256]
                 float* __restrict__ out) {         // [B, 256, HW]
    __shared__ float A_lds[2][MBLK][KSTR];
    __shared__ float B_lds[2][KSTAGE][KSTR];

    int p0     = blockIdx.x * NBLK;                 // pixel base (within batch)
    int b      = blockIdx.y;                        // batch
    int m_base = blockIdx.z * MBLK;                 // output-channel base

    int tid    = threadIdx.x;
    int lane   = tid & 31;
    int wv     = tid >> 5;
    int ntile  = wv & 3;
    int mgroup = wv >> 2;
    int l16    = lane & 15;
    int half   = lane >> 4;                         // K sub-pair select

    const float* fbase = fused + (size_t)b * CCH * HW;   // uniform SGPR base

    v8f acc[4] = {};

    // ---- async stage issue: buffer `buf`, K offset `k0` ----
    auto stage = [&](int buf, int k0) {
        // A: 128x64 f32 = 2048 float4 slots, 8 per thread
#pragma unroll
        for (int j = 0; j < 8; ++j) {
            int s   = tid + j * 256;
            int row = s >> 4;
            int k4  = (s & 15) << 2;
            async_ld_b128(lds_off32(&A_lds[buf][row][k4]),
                          (unsigned)(((m_base + row) * CCH + k0 + k4) * 4),
                          out_w);
        }
        // B: 64x64 f32 = 1024 float4 slots, 4 per thread
#pragma unroll
        for (int j = 0; j < 4; ++j) {
            int s   = tid + j * 256;
            int row = s >> 4;
            int n4  = (s & 15) << 2;
            async_ld_b128(lds_off32(&B_lds[buf][row][n4]),
                          (unsigned)(((k0 + row) * HW + p0 + n4) * 4),
                          fbase);
        }
    };

    stage(0, 0);                                    // prologue

    const int NSTAGE = CCH / KSTAGE;                // 4
    for (int s = 0; s < NSTAGE; ++s) {
        int cur = s & 1;
        wait_async0();                              // our DMA into `cur` done
        __syncthreads();                            // everyone's DMA done;
                                                    // all reads of `cur^1`
                                                    // from iter s-1 retired
        if (s + 1 < NSTAGE)
            stage(cur ^ 1, (s + 1) * KSTAGE);       // overlap DMA with WMMAs

        const float (*A)[KSTR] = A_lds[cur];
        const float (*B)[KSTR] = B_lds[cur];
#pragma unroll
        for (int kk = 0; kk < KSTAGE; kk += 4) {
            int k2 = kk + half * 2;                 // lanes 0-15: K0/K1; 16-31: K2/K3
            v2f bf = { B[k2][ntile * 16 + l16],
                       B[k2 + 1][ntile * 16 + l16] };
#pragma unroll
            for (int t = 0; t < 4; ++t) {
                int mr = mgroup * 64 + t * 16 + l16;
                v2f af = { A[mr][k2], A[mr][k2 + 1] };
                acc[t] = __builtin_amdgcn_wmma_f32_16x16x4_f32(
                    /*neg_a=*/false, af, /*neg_b=*/false, bf,
                    /*c_mod=*/(short)0, acc[t],
                    /*reuse_a=*/false, /*reuse_b=*/false);
            }
        }
    }

    // D layout: VGPR r, lanes 0-15 -> M=r, lanes 16-31 -> M=r+8; N = lane%16
    int n = p0 + ntile * 16 + l16;
#pragma unroll
    for (int t = 0; t < 4; ++t) {
        int o0 = m_base + mgroup * 64 + t * 16 + half * 8;
#pragma unroll
        for (int r = 0; r < 8; ++r) {
            int o = o0 + r;
            out[((size_t)b * CCH + o) * HW + n] = acc[t][r] + out_b[o];
        }
    }
}

// ---------------------------------------------------------------- launcher --
extern "C" void kernel_launch(void* const* d_in, const int* in_sizes, int n_in,
                              void* d_out, int out_size, void* d_ws, size_t ws_size,
                              hipStream_t stream) {
    (void)in_sizes; (void)n_in; (void)out_size; (void)ws_size;
    const float* x      = (const float*)d_in[0];
    const float* dw_w   = (const float*)d_in[1];
    const float* dw_b   = (const float*)d_in[2];
    const float* attn_w = (const float*)d_in[3];
    const float* attn_b = (const float*)d_in[4];
    const float* out_w  = (const float*)d_in[5];
    const float* out_b  = (const float*)d_in[6];
    float* out = (float*)d_out;

    float* scale = (float*)d_ws;                                   // 3*NPIX f32
    float* fused = (float*)((char*)d_ws + (size_t)3 * NPIX * 4);   // B*C*HW f32

    attn_kernel<<<NPIX / 256, 256, 0, stream>>>(x, attn_w, attn_b, scale);
    fused_kernel<<<16 * CCH, 256, 0, stream>>>(x, dw_w, dw_b, scale, fused);
    gemm_kernel<<<dim3(HW / NBLK, 16, CCH / MBLK), 256, 0, stream>>>(fused, out_w, out_b, out);
}